// GCN_46471546142842
// MI455X (gfx1250) — compile-verified
//
#include <hip/hip_runtime.h>
#include <hip/hip_bf16.h>

#define N_NODES 10000
#define N_EDGES 100000
#define DIM     768

typedef __attribute__((ext_vector_type(2))) float v2f;
typedef __attribute__((ext_vector_type(8))) float v8f;

// ---------------------------------------------------------------------------
// 1) deg[i] = 1.0 (self loop)
// ---------------------------------------------------------------------------
__global__ void gcn_init_deg(float* __restrict__ deg) {
    int i = blockIdx.x * blockDim.x + threadIdx.x;
    if (i < N_NODES) deg[i] = 1.0f;
}

// ---------------------------------------------------------------------------
// 2) deg[dst[e]] += 1
// ---------------------------------------------------------------------------
__global__ void gcn_count_deg(const int* __restrict__ dst, float* __restrict__ deg) {
    int e = blockIdx.x * blockDim.x + threadIdx.x;
    if (e < N_EDGES) atomicAdd(&deg[dst[e]], 1.0f);
}

// ---------------------------------------------------------------------------
// 3) dinv[i] = rsqrt(deg[i])   (deg >= 1 always, self-loop guarantees it)
// ---------------------------------------------------------------------------
__global__ void gcn_rsqrt_deg(const float* __restrict__ deg, float* __restrict__ dinv) {
    int i = blockIdx.x * blockDim.x + threadIdx.x;
    if (i < N_NODES) dinv[i] = rsqrtf(deg[i]);
}

// ---------------------------------------------------------------------------
// 4) h = x @ W  via V_WMMA_F32_16X16X4_F32 (full fp32 precision).
//    One wave32 per 16(M) x 64(N) output strip; A fragment reused over 4
//    N-tiles per k-step. Fragment layouts per CDNA5 ISA 7.12.2 (fp32):
//      A 16x4 : lane<16 -> K=0,1 ; lane>=16 -> K=2,3 ; M = lane&15
//      B 4x16 : VGPR v  -> K = koff+v ; N = lane&15
//      C 16x16: VGPR r  -> M = r + (lane>=16)*8 ; N = lane&15
// ---------------------------------------------------------------------------
__global__ __launch_bounds__(32) void gcn_gemm_wmma(const float* __restrict__ X,
                                                    const float* __restrict__ W,
                                                    float* __restrict__ H) {
    const int lane = threadIdx.x & 31;
    const int m    = lane & 15;
    const int hi   = lane >> 4;       // 0 or 1
    const int koff = hi * 2;          // K sub-offset for A/B fragments
    const int row0 = blockIdx.x * 16; // M tile base
    const int col0 = blockIdx.y * 64; // N strip base (4 tiles)

    v8f c0 = {}, c1 = {}, c2 = {}, c3 = {};

    const float* xrow = X + (size_t)(row0 + m) * DIM;

    for (int k = 0; k < DIM; k += 4) {
        // A fragment: two consecutive f32 along K
        v2f a = *(const v2f*)(xrow + k + koff);

        // B fragments for the 4 N-tiles: rows K = k+koff, k+koff+1 at col m
        const float* wb = W + (size_t)(k + koff) * DIM + col0 + m;
        v2f b0 = { wb[ 0], wb[DIM +  0] };
        v2f b1 = { wb[16], wb[DIM + 16] };
        v2f b2 = { wb[32], wb[DIM + 32] };
        v2f b3 = { wb[48], wb[DIM + 48] };

        c0 = __builtin_amdgcn_wmma_f32_16x16x4_f32(false, a, false, b0, (short)0, c0, false, false);
        c1 = __builtin_amdgcn_wmma_f32_16x16x4_f32(false, a, false, b1, (short)0, c1, false, false);
        c2 = __builtin_amdgcn_wmma_f32_16x16x4_f32(false, a, false, b2, (short)0, c2, false, false);
        c3 = __builtin_amdgcn_wmma_f32_16x16x4_f32(false, a, false, b3, (short)0, c3, false, false);
    }

    // Store D fragments
    #pragma unroll
    for (int r = 0; r < 8; ++r) {
        const int row = row0 + r + hi * 8;
        float* ho = H + (size_t)row * DIM + col0 + m;
        ho[ 0] = c0[r];
        ho[16] = c1[r];
        ho[32] = c2[r];
        ho[48] = c3[r];
    }
}

// ---------------------------------------------------------------------------
// 5) out[i] = b + dinv[i]^2 * h[i]   (self-loop message + bias; full overwrite
//    of the poisoned output buffer)
// ---------------------------------------------------------------------------
__global__ void gcn_self_bias(const float* __restrict__ H,
                              const float* __restrict__ dinv,
                              const float* __restrict__ bias,
                              float* __restrict__ out) {
    const int i = blockIdx.x;
    const float n = dinv[i] * dinv[i];
    const size_t base = (size_t)i * DIM;
    for (int d = threadIdx.x; d < DIM; d += blockDim.x)
        out[base + d] = bias[d] + n * H[base + d];
}

// ---------------------------------------------------------------------------
// 6) out[dst[e]] += dinv[src]*dinv[dst] * h[src[e]]   (one block per edge)
// ---------------------------------------------------------------------------
__global__ void gcn_scatter(const int* __restrict__ src,
                            const int* __restrict__ dst,
                            const float* __restrict__ H,
                            const float* __restrict__ dinv,
                            float* __restrict__ out) {
    const int e = blockIdx.x;
    const int s = src[e];
    const int t = dst[e];
    const float n = dinv[s] * dinv[t];
    const size_t sb = (size_t)s * DIM;
    const size_t tb = (size_t)t * DIM;
    for (int d = threadIdx.x; d < DIM; d += blockDim.x)
        atomicAdd(&out[tb + d], n * H[sb + d]);
}

// ---------------------------------------------------------------------------
extern "C" void kernel_launch(void* const* d_in, const int* in_sizes, int n_in,
                              void* d_out, int out_size, void* d_ws, size_t ws_size,
                              hipStream_t stream) {
    const float* x    = (const float*)d_in[0];
    const int*   ei   = (const int*)d_in[1];   // [2, N_EDGES]: row0=src, row1=dst
    const float* W    = (const float*)d_in[2];
    const float* bias = (const float*)d_in[3];
    float*       out  = (float*)d_out;

    const int* src = ei;
    const int* dst = ei + N_EDGES;

    float* deg  = (float*)d_ws;
    float* dinv = deg + N_NODES;
    float* h    = dinv + N_NODES;   // N_NODES*DIM floats (~30.7 MB)

    gcn_init_deg <<<(N_NODES + 255) / 256, 256, 0, stream>>>(deg);
    gcn_count_deg<<<(N_EDGES + 255) / 256, 256, 0, stream>>>(dst, deg);
    gcn_rsqrt_deg<<<(N_NODES + 255) / 256, 256, 0, stream>>>(deg, dinv);

    dim3 ggrid(N_NODES / 16, DIM / 64);   // 625 x 12 waves
    gcn_gemm_wmma<<<ggrid, 32, 0, stream>>>(x, W, h);

    gcn_self_bias<<<N_NODES, 256, 0, stream>>>(h, dinv, bias, out);
    gcn_scatter  <<<N_EDGES, 256, 0, stream>>>(src, dst, h, dinv, out);
}